// EncoderLayer_11003706212667
// MI455X (gfx1250) — compile-verified
//
#include <hip/hip_runtime.h>
#include <math.h>

// ---------------------------------------------------------------------------
// Transformer encoder layer for MI455X (gfx1250), fp32 via V_WMMA_F32_16X16X4_F32.
// B=4, S=2048, D=1024, F=4096.  Compute-bound (206 GFLOP vs ~100MB HBM traffic);
// h + weights are L2-resident (192MB).  Per-wave register-resident accumulators,
// LDS only for cross-wave tiles (P/U, padded to avoid bank conflicts) and stats.
// Async-to-LDS (global_load_async_to_lds_b32 + s_wait_asynccnt) stages the
// per-chunk broadcast operands (mask / b1); global_prefetch warms the next chunk.
// ---------------------------------------------------------------------------

typedef float v2f __attribute__((ext_vector_type(2)));
typedef float v8f __attribute__((ext_vector_type(8)));

constexpr int Bb = 4, Ss = 2048, Dd = 1024, Ff = 4096;
constexpr int BQ = 16;       // q rows per workgroup
constexpr int NW = 8;        // waves per workgroup (wave32)
constexpr int CH = 256;      // k / f columns per outer iteration (2 tiles/wave)
constexpr int PST = CH + 2;  // padded LDS row stride (dwords) -> distinct banks

__device__ __forceinline__ v8f wmma4(v2f a, v2f b, v8f c) {
  // D = A(16x4,f32) * B(4x16,f32) + C(16x16,f32)
  return __builtin_amdgcn_wmma_f32_16x16x4_f32(false, a, false, b, (short)0, c,
                                               false, false);
}

// CDNA5 async copy: global -> LDS, tracked by ASYNCcnt
__device__ __forceinline__ void async_ld_f32(unsigned lds_off,
                                             const float* gptr) {
  asm volatile("global_load_async_to_lds_b32 %0, %1, off"
               :
               : "v"(lds_off), "v"(gptr)
               : "memory");
}
__device__ __forceinline__ void wait_async0() {
  asm volatile("s_wait_asynccnt 0x0" ::: "memory");
}

// reductions across the 16-lane half-wave (matches C-matrix N striping)
__device__ __forceinline__ float half16_max(float v) {
#pragma unroll
  for (int o = 1; o < 16; o <<= 1) v = fmaxf(v, __shfl_xor(v, o, 32));
  return v;
}
__device__ __forceinline__ float half16_sum(float v) {
#pragma unroll
  for (int o = 1; o < 16; o <<= 1) v += __shfl_xor(v, o, 32);
  return v;
}

// ===========================================================================
// Kernel 1: flash attention (Q=K=V=h) + residual + LayerNorm1  ->  h1 (ws)
// ===========================================================================
__global__ __launch_bounds__(256) void attn_ln1_kernel(
    const float* __restrict__ h, const float* __restrict__ amask,
    const float* __restrict__ g1, const float* __restrict__ bt1,
    float* __restrict__ h1) {
  __shared__ float Plds[BQ * PST];  // ~16.1 KB probability tile (padded)
  __shared__ float mlds[CH];        // async-staged mask chunk
  __shared__ float wred[NW][16];
  __shared__ float wsq[NW][16];
  __shared__ float mrun[16], srun[16], resc[16], nmax[16], statA[16], statB[16];

  const int b = blockIdx.y;
  const int q0 = blockIdx.x * BQ;
  const int tid = threadIdx.x;
  const int wave = tid >> 5;
  const int lane = tid & 31;
  const int hlf = lane >> 4;
  const int l16 = lane & 15;
  const int kb = hlf << 1;  // K offset of this lane's fragment pair
  const long baseB = (long)b * Ss;

  if (tid < 16) { mrun[tid] = -INFINITY; srun[tid] = 0.f; }
  __syncthreads();

  v8f acc[8];  // O accumulator: 16 rows x 128 cols (this wave's D slice)
#pragma unroll
  for (int t = 0; t < 8; ++t) acc[t] = (v8f){0.f, 0.f, 0.f, 0.f, 0.f, 0.f, 0.f, 0.f};

  const float* qrow = h + (baseB + q0 + l16) * Dd;
  const float scale = 0.03125f;  // 1/sqrt(1024)

  for (int j = 0; j < Ss; j += CH) {
    // ---- stage mask chunk via async-to-LDS; prefetch next K block ----
    async_ld_f32((unsigned)(uintptr_t)&mlds[tid], amask + baseB + j + tid);
    wait_async0();
    if (j + CH < Ss) __builtin_prefetch(h + (baseB + j + CH + tid) * Dd, 0, 1);
    __syncthreads();

    // ---- 1) two 16x16 score tiles, A-fragment shared ----
    const float* krow0 = h + (baseB + j + wave * 32 + l16) * Dd;
    const float* krow1 = krow0 + (long)16 * Dd;
    v8f c0 = (v8f){0.f, 0.f, 0.f, 0.f, 0.f, 0.f, 0.f, 0.f};
    v8f c1 = c0;
    for (int kk = 0; kk < Dd; kk += 4) {
      v2f a = *(const v2f*)(qrow + kk + kb);
      v2f bb0 = *(const v2f*)(krow0 + kk + kb);
      v2f bb1 = *(const v2f*)(krow1 + kk + kb);
      c0 = wmma4(a, bb0, c0);
      c1 = wmma4(a, bb1, c1);
    }
    const float mv0 = mlds[wave * 32 + l16];
    const float mv1 = mlds[wave * 32 + 16 + l16];
    float sc0[8], sc1[8];
#pragma unroll
    for (int e = 0; e < 8; ++e) {
      sc0[e] = c0[e] * scale + mv0;
      sc1[e] = c1[e] * scale + mv1;
    }

    // ---- 2) per-wave row max over 32 columns -> LDS ----
#pragma unroll
    for (int e = 0; e < 8; ++e) {
      float v = fmaxf(half16_max(sc0[e]), half16_max(sc1[e]));
      if (l16 == 0) wred[wave][e + 8 * hlf] = v;
    }
    __syncthreads();

    // ---- 3) online-softmax running max / rescale factors ----
    if (tid < 16) {
      float tm = wred[0][tid];
#pragma unroll
      for (int w = 1; w < NW; ++w) tm = fmaxf(tm, wred[w][tid]);
      float mo = mrun[tid];
      float nm = fmaxf(mo, tm);
      nmax[tid] = nm;
      resc[tid] = (mo > -1e30f) ? expf(mo - nm) : 0.f;
      mrun[tid] = nm;
    }
    __syncthreads();

    // ---- 4) P = exp(sc - m), stage P, rescale O, partial row sums ----
    float nmv[8], rs[8];
#pragma unroll
    for (int e = 0; e < 8; ++e) {
      nmv[e] = nmax[e + 8 * hlf];
      rs[e] = resc[e + 8 * hlf];
    }
#pragma unroll
    for (int e = 0; e < 8; ++e) {
      float p0 = expf(sc0[e] - nmv[e]);
      float p1 = expf(sc1[e] - nmv[e]);
      sc0[e] = p0;
      sc1[e] = p1;
      Plds[(e + 8 * hlf) * PST + wave * 32 + l16] = p0;
      Plds[(e + 8 * hlf) * PST + wave * 32 + 16 + l16] = p1;
    }
#pragma unroll
    for (int t = 0; t < 8; ++t)
#pragma unroll
      for (int e = 0; e < 8; ++e) acc[t][e] *= rs[e];
#pragma unroll
    for (int e = 0; e < 8; ++e) {
      float v = half16_sum(sc0[e]) + half16_sum(sc1[e]);
      if (l16 == 0) wred[wave][e + 8 * hlf] = v;
    }
    __syncthreads();

    if (tid < 16) {
      float sm = 0.f;
#pragma unroll
      for (int w = 0; w < NW; ++w) sm += wred[w][tid];
      srun[tid] = srun[tid] * resc[tid] + sm;
    }

    // ---- 5) O += P @ V : kk-outer so one A-fragment feeds 8 WMMAs ----
    for (int kk = 0; kk < CH; kk += 4) {
      v2f a = *(const v2f*)(&Plds[l16 * PST + kk + kb]);
      const float* vr0 = h + (baseB + j + kk + kb) * Dd + wave * 128 + l16;
      const float* vr1 = vr0 + Dd;
#pragma unroll
      for (int t = 0; t < 8; ++t) {
        v2f bb;
        bb.x = vr0[t * 16];
        bb.y = vr1[t * 16];
        acc[t] = wmma4(a, bb, acc[t]);
      }
    }
    __syncthreads();  // protect Plds / mlds / wred before next iteration
  }

  // ---- normalize by row sum, add residual, LayerNorm1 ----
  if (tid < 16) statA[tid] = 1.f / srun[tid];
  __syncthreads();

  float si[8];
#pragma unroll
  for (int e = 0; e < 8; ++e) si[e] = statA[e + 8 * hlf];

  float ps[8], pq[8];
#pragma unroll
  for (int e = 0; e < 8; ++e) { ps[e] = 0.f; pq[e] = 0.f; }
#pragma unroll
  for (int t = 0; t < 8; ++t) {
#pragma unroll
    for (int e = 0; e < 8; ++e) {
      int row = e + 8 * hlf;
      int d = wave * 128 + t * 16 + l16;
      float v = acc[t][e] * si[e] + h[(baseB + q0 + row) * Dd + d];
      acc[t][e] = v;
      ps[e] += v;
      pq[e] += v * v;
    }
  }
#pragma unroll
  for (int e = 0; e < 8; ++e) {
    float a = half16_sum(ps[e]);
    float q = half16_sum(pq[e]);
    if (l16 == 0) { wred[wave][e + 8 * hlf] = a; wsq[wave][e + 8 * hlf] = q; }
  }
  __syncthreads();
  if (tid < 16) {
    float sm = 0.f, sq = 0.f;
#pragma unroll
    for (int w = 0; w < NW; ++w) { sm += wred[w][tid]; sq += wsq[w][tid]; }
    float mu = sm * (1.f / Dd);
    float var = sq * (1.f / Dd) - mu * mu;
    statA[tid] = mu;
    statB[tid] = rsqrtf(var + 1e-5f);
  }
  __syncthreads();
  float muv[8], rv[8];
#pragma unroll
  for (int e = 0; e < 8; ++e) { muv[e] = statA[e + 8 * hlf]; rv[e] = statB[e + 8 * hlf]; }
#pragma unroll
  for (int t = 0; t < 8; ++t) {
#pragma unroll
    for (int e = 0; e < 8; ++e) {
      int row = e + 8 * hlf;
      int d = wave * 128 + t * 16 + l16;
      float y = (acc[t][e] - muv[e]) * rv[e] * g1[d] + bt1[d];
      h1[(baseB + q0 + row) * Dd + d] = y;
    }
  }
}

// ===========================================================================
// Kernel 2: fused FFN (h1@w1+b1 -> exact GELU -> @w2+b2) + residual + LN2 -> out
// ===========================================================================
__global__ __launch_bounds__(256) void ffn_ln2_kernel(
    const float* __restrict__ h1, const float* __restrict__ w1,
    const float* __restrict__ b1, const float* __restrict__ w2,
    const float* __restrict__ b2, const float* __restrict__ g2,
    const float* __restrict__ bt2, float* __restrict__ out) {
  __shared__ float Ulds[BQ * PST];  // ~16.1 KB GELU-activation tile (padded)
  __shared__ float blds[CH];        // async-staged b1 chunk
  __shared__ float wred[NW][16];
  __shared__ float wsq[NW][16];
  __shared__ float statA[16], statB[16];

  const int b = blockIdx.y;
  const int q0 = blockIdx.x * BQ;
  const int tid = threadIdx.x;
  const int wave = tid >> 5;
  const int lane = tid & 31;
  const int hlf = lane >> 4;
  const int l16 = lane & 15;
  const int kb = hlf << 1;
  const long baseB = (long)b * Ss;

  v8f acc[8];  // FFN output accumulator: 16 rows x this wave's 128-wide D slice
#pragma unroll
  for (int t = 0; t < 8; ++t) acc[t] = (v8f){0.f, 0.f, 0.f, 0.f, 0.f, 0.f, 0.f, 0.f};

  const float* arow = h1 + (baseB + q0 + l16) * Dd;

  for (int f0 = 0; f0 < Ff; f0 += CH) {
    // ---- stage b1 chunk via async-to-LDS; prefetch next w2 rows ----
    async_ld_f32((unsigned)(uintptr_t)&blds[tid], b1 + f0 + tid);
    wait_async0();
    if (f0 + CH < Ff)
      __builtin_prefetch(w2 + (long)(f0 + CH + tid) * Dd, 0, 1);
    __syncthreads();

    // ---- two U tiles: GELU(h1 @ w1 + b1), A-fragment shared ----
    const int fc0 = f0 + wave * 32 + l16;
    const float* w1c0 = w1 + fc0;
    const float* w1c1 = w1 + fc0 + 16;
    v8f c0 = (v8f){0.f, 0.f, 0.f, 0.f, 0.f, 0.f, 0.f, 0.f};
    v8f c1 = c0;
    for (int kk = 0; kk < Dd; kk += 4) {
      v2f a = *(const v2f*)(arow + kk + kb);
      const float* p0 = w1c0 + (long)(kk + kb) * Ff;
      const float* p1 = w1c1 + (long)(kk + kb) * Ff;
      v2f bb0; bb0.x = p0[0]; bb0.y = p0[Ff];
      v2f bb1; bb1.x = p1[0]; bb1.y = p1[Ff];
      c0 = wmma4(a, bb0, c0);
      c1 = wmma4(a, bb1, c1);
    }
    const float bias0 = blds[wave * 32 + l16];
    const float bias1 = blds[wave * 32 + 16 + l16];
#pragma unroll
    for (int e = 0; e < 8; ++e) {
      float x0 = c0[e] + bias0;
      float x1 = c1[e] + bias1;
      float g0 = 0.5f * x0 * (1.f + erff(x0 * 0.70710678118654752f));
      float g1v = 0.5f * x1 * (1.f + erff(x1 * 0.70710678118654752f));
      Ulds[(e + 8 * hlf) * PST + wave * 32 + l16] = g0;
      Ulds[(e + 8 * hlf) * PST + wave * 32 + 16 + l16] = g1v;
    }
    __syncthreads();

    // ---- acc += U @ w2 : kk-outer so one A-fragment feeds 8 WMMAs ----
    for (int kk = 0; kk < CH; kk += 4) {
      v2f a = *(const v2f*)(&Ulds[l16 * PST + kk + kb]);
      const float* wr0 = w2 + (long)(f0 + kk + kb) * Dd + wave * 128 + l16;
      const float* wr1 = wr0 + Dd;
#pragma unroll
      for (int t = 0; t < 8; ++t) {
        v2f bb;
        bb.x = wr0[t * 16];
        bb.y = wr1[t * 16];
        acc[t] = wmma4(a, bb, acc[t]);
      }
    }
    __syncthreads();  // protect Ulds / blds before next f-chunk
  }

  // ---- + b2 + residual(h1), LayerNorm2 -> out ----
  float ps[8], pq[8];
#pragma unroll
  for (int e = 0; e < 8; ++e) { ps[e] = 0.f; pq[e] = 0.f; }
#pragma unroll
  for (int t = 0; t < 8; ++t) {
#pragma unroll
    for (int e = 0; e < 8; ++e) {
      int row = e + 8 * hlf;
      int d = wave * 128 + t * 16 + l16;
      float v = acc[t][e] + b2[d] + h1[(baseB + q0 + row) * Dd + d];
      acc[t][e] = v;
      ps[e] += v;
      pq[e] += v * v;
    }
  }
#pragma unroll
  for (int e = 0; e < 8; ++e) {
    float a = half16_sum(ps[e]);
    float q = half16_sum(pq[e]);
    if (l16 == 0) { wred[wave][e + 8 * hlf] = a; wsq[wave][e + 8 * hlf] = q; }
  }
  __syncthreads();
  if (tid < 16) {
    float sm = 0.f, sq = 0.f;
#pragma unroll
    for (int w = 0; w < NW; ++w) { sm += wred[w][tid]; sq += wsq[w][tid]; }
    float mu = sm * (1.f / Dd);
    float var = sq * (1.f / Dd) - mu * mu;
    statA[tid] = mu;
    statB[tid] = rsqrtf(var + 1e-5f);
  }
  __syncthreads();
  float muv[8], rv[8];
#pragma unroll
  for (int e = 0; e < 8; ++e) { muv[e] = statA[e + 8 * hlf]; rv[e] = statB[e + 8 * hlf]; }
#pragma unroll
  for (int t = 0; t < 8; ++t) {
#pragma unroll
    for (int e = 0; e < 8; ++e) {
      int row = e + 8 * hlf;
      int d = wave * 128 + t * 16 + l16;
      float y = (acc[t][e] - muv[e]) * rv[e] * g2[d] + bt2[d];
      out[(baseB + q0 + row) * Dd + d] = y;
    }
  }
}

// ===========================================================================
extern "C" void kernel_launch(void* const* d_in, const int* in_sizes, int n_in,
                              void* d_out, int out_size, void* d_ws,
                              size_t ws_size, hipStream_t stream) {
  (void)in_sizes; (void)n_in; (void)out_size; (void)ws_size;
  const float* h    = (const float*)d_in[0];  // [B,S,D]
  const float* mask = (const float*)d_in[1];  // [B,S]
  const float* w1   = (const float*)d_in[2];  // [D,F]
  const float* b1   = (const float*)d_in[3];  // [F]
  const float* w2   = (const float*)d_in[4];  // [F,D]
  const float* b2   = (const float*)d_in[5];  // [D]
  const float* g1   = (const float*)d_in[6];
  const float* bt1  = (const float*)d_in[7];
  const float* g2   = (const float*)d_in[8];
  const float* bt2  = (const float*)d_in[9];
  float* h1  = (float*)d_ws;   // [B,S,D] = 32 MB scratch
  float* out = (float*)d_out;  // [B,S,D]

  dim3 grid(Ss / BQ, Bb);  // 128 x 4 = 512 workgroups
  attn_ln1_kernel<<<grid, 256, 0, stream>>>(h, mask, g1, bt1, h1);
  ffn_ln2_kernel<<<grid, 256, 0, stream>>>(h1, w1, b1, w2, b2, g2, bt2, out);
}